// Net_79946521248007
// MI455X (gfx1250) — compile-verified
//
#include <hip/hip_runtime.h>
#include <hip/hip_bf16.h>

// ---------------------------------------------------------------------------
// Gaussian-mixture CNN forward for MI455X (gfx1250, wave32).
//
// Hot loop: md[k,n] = (p_k-p_n)^T Ci_n (p_k-p_n) rewritten as rank-6 GEMM
//   md = A(Npad x 8) * B(8 x Npad),  A[k]=[x^2,xy,y^2,x,y,1,0,0]
// computed with V_WMMA_F32_16X16X4_F32 (two k=4 steps per 16x16 tile),
// fused with w_n * exp(-md/2) and per-row reduction via wave32 shfl_xor.
// All per-(b,lo) intermediate state lives in LDS (~66 KB << 320 KB/WGP).
// Workspace use: ~56K floats (~220 KB) for tot scalars + stage outputs.
// ---------------------------------------------------------------------------

typedef float v2f __attribute__((ext_vector_type(2)));
typedef float v8f __attribute__((ext_vector_type(8)));

#define NPAD_MAX 640        // max padded N (layer2: 625 -> 640)
#define TPB      256        // 8 wave32 waves per workgroup
#define NWAVES   (TPB/32)

__device__ __constant__ const float kTWO_PI = 6.28318530717958647692f;
#define EPS_F 1e-6f

__device__ __forceinline__ float det2f(float c00, float c01, float c10, float c11) {
    return c00 * c11 - c01 * c10;
}

// batchnorm (per_gaussian=True) normalizer: tot[b] = sum_d |w*2pi*sqrt(max(det,eps))|
__global__ void gm_tot_batch(const float* __restrict__ x, float* __restrict__ tot, int Nd) {
    int b = threadIdx.x;
    if (b >= 32) return;
    float s = 0.f;
    for (int d = 0; d < Nd; ++d) {
        const float* g = x + (b * Nd + d) * 7;
        float dt = det2f(g[3], g[4], g[5], g[6]);
        s += fabsf(g[0] * kTWO_PI * sqrtf(fmaxf(dt, EPS_F)));
    }
    tot[b] = s;
}

// batchnorm (per_gaussian=False): tot[li] = mean_b sum_n |integral|  (x: B x Lprev x Nsel x 7)
__global__ void gm_tot_chan(const float* __restrict__ x, float* __restrict__ tot,
                            int Lprev, int Nsel) {
    int li = threadIdx.x;
    if (li >= Lprev) return;
    float s = 0.f;
    for (int b = 0; b < 32; ++b)
        for (int n = 0; n < Nsel; ++n) {
            const float* g = x + ((b * Lprev + li) * Nsel + n) * 7;
            float dt = det2f(g[3], g[4], g[5], g[6]);
            s += fabsf(g[0] * kTWO_PI * sqrtf(fmaxf(dt, EPS_F)));
        }
    tot[li] = s * (1.0f / 32.0f);
}

// Fused: batchnorm(w /= tot) + gm_convolve + relu_fit (eval_at_centers via WMMA,
// relu scaling, top-Nout selection). One workgroup per (b, lo).
__global__ __launch_bounds__(TPB) void gm_layer(
    const float* __restrict__ xprev,   // (B, Li, Nd, 7)
    const float* __restrict__ kern,    // (Lo, Li, Nk, 7)
    const float* __restrict__ bias,    // (Lo)
    const float* __restrict__ tot,     // (B) if perBatch else (Li)
    float* __restrict__ xnext,         // (B, Lo, Nout, 7)
    int Li, int Nd, int Nk, int Lo, int Nout, int perBatch)
{
    __shared__ float Af[NPAD_MAX][8];      // k-side features (row-major)
    __shared__ float Bf[8][NPAD_MAX];      // n-side coefficients
    __shared__ float wArr[NPAD_MAX], vArr[NPAD_MAX];
    __shared__ float w2A[NPAD_MAX], keyA[NPAD_MAX];
    __shared__ float pxA[NPAD_MAX], pyA[NPAD_MAX];
    __shared__ float c00A[NPAD_MAX], c01A[NPAD_MAX], c11A[NPAD_MAX];
    __shared__ float redV[TPB];
    __shared__ int   redI[TPB];

    const int tid = threadIdx.x;
    const int b   = blockIdx.x / Lo;
    const int lo  = blockIdx.x % Lo;
    const int N    = Li * Nd * Nk;
    const int Npad = (N + 15) & ~15;
    const int nT   = Npad >> 4;

    // ---- Phase A: fused batchnorm + gm_convolve + WMMA feature build --------
    for (int n = tid; n < Npad; n += TPB) {
        float w = 0.f, px = 0.f, py = 0.f;
        float cs00 = 0.f, cs01 = 0.f, cs11 = 0.f;
        float a0=0,a1=0,a2=0,a3=0,a4=0,a5=0;
        float b0=0,b1=0,b2=0,b3=0,b4=0,b5=0;
        if (n < N) {
            int li = n / (Nd * Nk);
            int r  = n - li * (Nd * Nk);
            int d  = r / Nk;
            int kk = r - d * Nk;
            const float* gd = xprev + ((b * Li + li) * Nd + d) * 7;
            const float* gk = kern + ((lo * Li + li) * Nk + kk) * 7;
            float tv = perBatch ? tot[b] : tot[li];
            float wd = gd[0] / (tv + EPS_F);                 // batchnorm on w only
            float dd = det2f(gd[3], gd[4], gd[5], gd[6]);
            float dk = det2f(gk[3], gk[4], gk[5], gk[6]);
            cs00 = gd[3] + gk[3]; cs01 = gd[4] + gk[4];
            float cs10 = gd[5] + gk[5]; cs11 = gd[6] + gk[6];
            float ds = cs00 * cs11 - cs01 * cs10;
            w  = wd * gk[0] * kTWO_PI * sqrtf(dd * dk / fmaxf(ds, EPS_F));
            px = gd[1] + gk[1]; py = gd[2] + gk[2];
            float inv_d = 1.0f / ds;                          // reference inv2: no eps
            float ia =  cs11 * inv_d;                         // Ci00
            float ib = -cs01 * inv_d;                         // Ci01 == Ci10 (symmetric)
            float ic =  cs00 * inv_d;                         // Ci11
            a0 = px * px; a1 = px * py; a2 = py * py;
            a3 = px;      a4 = py;      a5 = 1.0f;
            b0 = ia; b1 = 2.0f * ib; b2 = ic;
            b3 = -2.0f * (ia * px + ib * py);
            b4 = -2.0f * (ib * px + ic * py);
            b5 = ia * px * px + 2.0f * ib * px * py + ic * py * py;
        }
        Af[n][0]=a0; Af[n][1]=a1; Af[n][2]=a2; Af[n][3]=a3;
        Af[n][4]=a4; Af[n][5]=a5; Af[n][6]=0.f; Af[n][7]=0.f;
        Bf[0][n]=b0; Bf[1][n]=b1; Bf[2][n]=b2; Bf[3][n]=b3;
        Bf[4][n]=b4; Bf[5][n]=b5; Bf[6][n]=0.f; Bf[7][n]=0.f;
        wArr[n]=w; pxA[n]=px; pyA[n]=py;
        c00A[n]=cs00; c01A[n]=cs01; c11A[n]=cs11;
    }
    __syncthreads();

    // ---- Phase B: v[k] = sum_n w_n * exp(-0.5 * md[k,n]) via WMMA -----------
    // D/C layout (16x16 f32): VGPR j, lanes 0-15 -> row j, lanes 16-31 -> row j+8,
    // lane&15 -> column. A frag: lane&15 -> M, VGPR j -> K = (lane>>4)*2 + j.
    const int lane  = tid & 31;
    const int wv    = tid >> 5;
    const int mIdx  = lane & 15;
    const int kHalf = (lane >> 4) * 2;
    for (int kt = wv; kt < nT; kt += NWAVES) {
        const int rowBase = kt * 16;
        v2f aLo, aHi;
        aLo.x = Af[rowBase + mIdx][kHalf + 0];
        aLo.y = Af[rowBase + mIdx][kHalf + 1];
        aHi.x = Af[rowBase + mIdx][4 + kHalf + 0];
        aHi.y = Af[rowBase + mIdx][4 + kHalf + 1];
        float acc[8];
        #pragma unroll
        for (int j = 0; j < 8; ++j) acc[j] = 0.f;
        for (int nt = 0; nt < nT; ++nt) {
            const int col = nt * 16 + mIdx;
            v2f bLo, bHi;
            bLo.x = Bf[kHalf + 0][col];
            bLo.y = Bf[kHalf + 1][col];
            bHi.x = Bf[4 + kHalf + 0][col];
            bHi.y = Bf[4 + kHalf + 1][col];
            v8f c = {0.f,0.f,0.f,0.f,0.f,0.f,0.f,0.f};
            c = __builtin_amdgcn_wmma_f32_16x16x4_f32(false, aLo, false, bLo,
                                                      (short)0, c, false, false);
            c = __builtin_amdgcn_wmma_f32_16x16x4_f32(false, aHi, false, bHi,
                                                      (short)0, c, false, false);
            const float wn = wArr[col];
            #pragma unroll
            for (int j = 0; j < 8; ++j)
                acc[j] += wn * __expf(-0.5f * c[j]);
        }
        #pragma unroll
        for (int j = 0; j < 8; ++j) {
            float s = acc[j];                    // row-sum across 16 columns
            s += __shfl_xor(s, 1, 32);
            s += __shfl_xor(s, 2, 32);
            s += __shfl_xor(s, 4, 32);
            s += __shfl_xor(s, 8, 32);
            if (mIdx == 0)
                vArr[rowBase + j + (lane >> 4) * 8] = s;
        }
    }
    __syncthreads();

    // ---- Phase C: bias + relu scaling --------------------------------------
    const float bl = bias[lo];
    for (int k = tid; k < Npad; k += TPB) {
        if (k < N) {
            float v = vArr[k] + bl;
            float scale = fmaxf(v, 0.f) / (fabsf(v) + EPS_F);
            float w2 = wArr[k] * scale;
            w2A[k]  = w2;
            keyA[k] = fabsf(w2);
        } else {
            w2A[k] = 0.f; keyA[k] = -1.0f;
        }
    }
    __syncthreads();

    // ---- Phase D: top-Nout by |w2| (iterative block argmax) ----------------
    for (int it = 0; it < Nout; ++it) {
        float bv = -2.f; int bi = 1 << 30;
        for (int k = tid; k < N; k += TPB) {
            float v = keyA[k];
            if (v > bv) { bv = v; bi = k; }
        }
        redV[tid] = bv; redI[tid] = bi;
        __syncthreads();
        for (int s = TPB / 2; s > 0; s >>= 1) {
            if (tid < s) {
                float ov = redV[tid + s]; int oi = redI[tid + s];
                if (ov > redV[tid] || (ov == redV[tid] && oi < redI[tid])) {
                    redV[tid] = ov; redI[tid] = oi;
                }
            }
            __syncthreads();
        }
        if (tid == 0) {
            int sel = redI[0];
            float* o = xnext + ((b * Lo + lo) * Nout + it) * 7;
            o[0] = w2A[sel]; o[1] = pxA[sel]; o[2] = pyA[sel];
            o[3] = c00A[sel]; o[4] = c01A[sel]; o[5] = c01A[sel]; o[6] = c11A[sel];
            keyA[sel] = -3.f;
        }
        __syncthreads();
    }
}

// Final: batchnorm (mean over batch) + integrate + log_softmax. x3: (32,10,5,7).
__global__ void gm_final(const float* __restrict__ x3, float* __restrict__ out) {
    __shared__ float Isum[320], Aabs[320], tot4[10], sv[320];
    const int t = threadIdx.x;
    if (t < 320) {
        float I = 0.f, A = 0.f;
        for (int n = 0; n < 5; ++n) {
            const float* g = x3 + (t * 5 + n) * 7;
            float c = g[0] * kTWO_PI * sqrtf(fmaxf(det2f(g[3], g[4], g[5], g[6]), EPS_F));
            I += c; A += fabsf(c);
        }
        Isum[t] = I; Aabs[t] = A;
    }
    __syncthreads();
    if (t < 10) {
        float s = 0.f;
        for (int b = 0; b < 32; ++b) s += Aabs[b * 10 + t];
        tot4[t] = s * (1.0f / 32.0f);
    }
    __syncthreads();
    if (t < 320) sv[t] = Isum[t] / (tot4[t % 10] + EPS_F);
    __syncthreads();
    if (t < 32) {
        float mx = -1e30f;
        for (int lo = 0; lo < 10; ++lo) mx = fmaxf(mx, sv[t * 10 + lo]);
        float se = 0.f;
        for (int lo = 0; lo < 10; ++lo) se += __expf(sv[t * 10 + lo] - mx);
        float lse = mx + logf(se);
        for (int lo = 0; lo < 10; ++lo) out[t * 10 + lo] = sv[t * 10 + lo] - lse;
    }
}

extern "C" void kernel_launch(void* const* d_in, const int* in_sizes, int n_in,
                              void* d_out, int out_size, void* d_ws, size_t ws_size,
                              hipStream_t stream) {
    const float* in_x = (const float*)d_in[0];   // (32,1,64,7)
    const float* k1   = (const float*)d_in[1];   // (5,1,5,7)
    const float* k2   = (const float*)d_in[2];   // (6,5,5,7)
    const float* k3   = (const float*)d_in[3];   // (10,6,5,7)
    const float* b1   = (const float*)d_in[4];   // (5)
    const float* b2   = (const float*)d_in[5];   // (6)
    const float* b3   = (const float*)d_in[6];   // (10)
    float* out = (float*)d_out;                  // (32,10)
    float* ws  = (float*)d_ws;

    float* tot1 = ws;                 // 32
    float* x1   = ws + 32;            // 32*5*25*7 = 28000
    float* tot2 = ws + 28032;         // 8
    float* x2   = ws + 28040;         // 32*6*12*7 = 16128
    float* tot3 = ws + 44168;         // 8
    float* x3   = ws + 44176;         // 32*10*5*7 = 11200  (end: 55376 floats)

    gm_tot_batch<<<1, 32, 0, stream>>>(in_x, tot1, 64);
    gm_layer<<<32 * 5, TPB, 0, stream>>>(in_x, k1, b1, tot1, x1, 1, 64, 5, 5, 25, 1);
    gm_tot_chan<<<1, 32, 0, stream>>>(x1, tot2, 5, 25);
    gm_layer<<<32 * 6, TPB, 0, stream>>>(x1, k2, b2, tot2, x2, 5, 25, 5, 6, 12, 0);
    gm_tot_chan<<<1, 32, 0, stream>>>(x2, tot3, 6, 12);
    gm_layer<<<32 * 10, TPB, 0, stream>>>(x2, k3, b3, tot3, x3, 6, 12, 5, 10, 5, 0);
    gm_final<<<1, 320, 0, stream>>>(x3, out);
}